// DGCF_10857677324434
// MI455X (gfx1250) — compile-verified
//
#include <hip/hip_runtime.h>
#include <math.h>

#define NUM_USER 60000
#define NUM_ITEM 40000
#define NNODES   100000
#define NEDGES   500000
#define DIM      64
#define KFAC     4
#define DK       16

typedef float v2f __attribute__((ext_vector_type(2)));
typedef float v8f __attribute__((ext_vector_type(8)));

// ---------------- utility kernels ----------------

__global__ void k_zero(float* __restrict__ p, int n) {
    int i = blockIdx.x * blockDim.x + threadIdx.x;
    if (i < n) p[i] = 0.0f;
}

__global__ void k_copy(const float* __restrict__ src, float* __restrict__ dst, int n) {
    int i = blockIdx.x * blockDim.x + threadIdx.x;
    if (i < n) dst[i] = src[i];
}

__global__ void k_deg(const int* __restrict__ row, const int* __restrict__ col,
                      float* __restrict__ deg) {
    int e = blockIdx.x * blockDim.x + threadIdx.x;
    if (e < NEDGES) {
        atomicAdd(&deg[row[e]], 1.0f);
        atomicAdd(&deg[col[e]], 1.0f);
    }
}

__global__ void k_dinv(float* __restrict__ deg) {
    int i = blockIdx.x * blockDim.x + threadIdx.x;
    if (i < NNODES) {
        float d = deg[i];
        deg[i] = (d > 0.0f) ? (1.0f / sqrtf(d)) : 0.0f;
    }
}

__global__ void k_init_ego(const float* __restrict__ ue, const float* __restrict__ ie,
                           float* __restrict__ ego, float* __restrict__ all) {
    int i = blockIdx.x * blockDim.x + threadIdx.x;
    if (i < NNODES * DIM) {
        float v = (i < NUM_USER * DIM) ? ue[i] : ie[i - NUM_USER * DIM];
        ego[i] = v;
        all[i] = v;
    }
}

__global__ void k_accum(float* __restrict__ all, const float* __restrict__ ego) {
    int i = blockIdx.x * blockDim.x + threadIdx.x;
    if (i < NNODES * DIM) all[i] += ego[i];
}

// ---------------- softmax over 4 factors per edge ----------------

__global__ void k_softmax(float* __restrict__ Sw) {
    int e = blockIdx.x * blockDim.x + threadIdx.x;
    if (e >= NEDGES) return;
    float s0 = Sw[0 * NEDGES + e], s1 = Sw[1 * NEDGES + e];
    float s2 = Sw[2 * NEDGES + e], s3 = Sw[3 * NEDGES + e];
    float m = fmaxf(fmaxf(s0, s1), fmaxf(s2, s3));
    float e0 = expf(s0 - m), e1 = expf(s1 - m), e2 = expf(s2 - m), e3 = expf(s3 - m);
    float inv = 1.0f / (e0 + e1 + e2 + e3);
    Sw[0 * NEDGES + e] = e0 * inv;
    Sw[1 * NEDGES + e] = e1 * inv;
    Sw[2 * NEDGES + e] = e2 * inv;
    Sw[3 * NEDGES + e] = e3 * inv;
}

// ---------------- conv: gather-scale-scatter over 2E symmetrized edges ----------------
// thread = (direction*E + e, dim)

__global__ void k_conv(const int* __restrict__ row, const int* __restrict__ col,
                       const float* __restrict__ dinv, const float* __restrict__ Sw,
                       const float* __restrict__ ego, float* __restrict__ xnew) {
    int t = blockIdx.x * blockDim.x + threadIdx.x;     // < 2*E*64 = 64M
    int d  = t & (DIM - 1);
    int e2 = t >> 6;
    int e  = (e2 < NEDGES) ? e2 : (e2 - NEDGES);
    int r  = (e2 < NEDGES) ? row[e] : col[e];
    int c  = (e2 < NEDGES) ? col[e] : row[e];
    float w = dinv[r] * dinv[c] * Sw[(d >> 4) * NEDGES + e];
    atomicAdd(&xnew[c * DIM + d], w * ego[r * DIM + d]);
}

// ---------------- routing: S[k][e] += <l2norm(xnew[row0]), tanh(l2norm(ego[col0]))> ----
// One wave per tile of (16 edges, 1 factor). Build U and tanh(V) in LDS, then chain
// 4x V_WMMA_F32_16X16X4_F32 (K=16 total) to get the 16x16 Gram matrix; its diagonal
// holds the 16 routing-score updates. Full f32 precision.

__global__ void __launch_bounds__(256)
k_routing(const int* __restrict__ row, const int* __restrict__ col,
          const float* __restrict__ ego, const float* __restrict__ xnew,
          float* __restrict__ Sw) {
    __shared__ float uT[8][16][17];   // [wave][edge][dim], padded vs bank conflicts
    __shared__ float vT[8][16][17];

    int lane = threadIdx.x & 31;
    int wv   = threadIdx.x >> 5;
    int tile = blockIdx.x * 8 + wv;   // tiles = (E/16)*KFAC = 125000, launched exactly
    int kf    = tile & (KFAC - 1);    // factor
    int ebase = (tile >> 2) * 16;     // first edge of the 16-edge group
    int half  = lane >> 4;            // 0 | 1
    int d     = lane & 15;            // dim (phase 1) / M=N index (phase 2)

    float (*u)[17]  = uT[wv];
    float (*tv)[17] = vT[wv];

    // Phase 1: per (edge, dim) compute normalized u and tanh(normalized v) -> LDS
    for (int p = 0; p < 8; ++p) {
        int el = p * 2 + half;
        int e  = ebase + el;
        int r  = row[e], c = col[e];
        float xu = xnew[r * DIM + kf * DK + d];
        float xv = ego [c * DIM + kf * DK + d];
        float su = xu * xu, sv = xv * xv;
        #pragma unroll
        for (int m = 1; m < 16; m <<= 1) {     // reduce within each 16-lane half
            su += __shfl_xor(su, m, 16);
            sv += __shfl_xor(sv, m, 16);
        }
        float iu = 1.0f / fmaxf(sqrtf(su), 1e-12f);
        float iv = 1.0f / fmaxf(sqrtf(sv), 1e-12f);
        u [el][d] = xu * iu;
        tv[el][d] = tanhf(xv * iv);
    }
    __syncthreads();

    // Phase 2: D(16x16) = U(16x16dims) * tanh(V)^T via 4 chained f32 WMMAs (K=4 each).
    // A layout (16x4): lane->M, VGPR0 holds K= (half?2:0), VGPR1 K= (half?3:1).
    // B layout (4x16): lane->N, same VGPR/half -> K mapping.
    v8f acc = {0.f, 0.f, 0.f, 0.f, 0.f, 0.f, 0.f, 0.f};
    #pragma unroll
    for (int ck = 0; ck < 4; ++ck) {
        int k0 = 4 * ck + (half ? 2 : 0);
        int k1 = k0 + 1;
        v2f a, b;
        a.x = u [d][k0];  a.y = u [d][k1];
        b.x = tv[d][k0];  b.y = tv[d][k1];
        acc = __builtin_amdgcn_wmma_f32_16x16x4_f32(
            /*neg_a=*/false, a, /*neg_b=*/false, b,
            /*c_mod=*/(short)0, acc, /*reuse_a=*/false, /*reuse_b=*/false);
    }
    __syncthreads();

    // Phase 3: spill C to LDS per ISA C/D layout, pick diagonal, update S.
    #pragma unroll
    for (int r = 0; r < 8; ++r) {
        u[r + (half ? 8 : 0)][d] = acc[r];    // (M = r + 8*half, N = d)
    }
    __syncthreads();
    if (lane < 16) {
        Sw[kf * NEDGES + ebase + lane] += u[lane][lane];
    }
}

// ---------------- host orchestration ----------------

extern "C" void kernel_launch(void* const* d_in, const int* in_sizes, int n_in,
                              void* d_out, int out_size, void* d_ws, size_t ws_size,
                              hipStream_t stream) {
    const float* user_emb = (const float*)d_in[0];
    const float* item_emb = (const float*)d_in[1];
    const float* S_in     = (const float*)d_in[2];
    const int*   edge     = (const int*)d_in[3];      // [2, E]
    const int* row = edge;
    const int* col = edge + NEDGES;

    float* all = (float*)d_out;                       // N*64 accumulated embeddings
    float* Sw  = (float*)d_out + NNODES * DIM;        // KFAC*E working + final S

    float* ws   = (float*)d_ws;
    float* dinv = ws;                                 // N
    float* bufA = ws + NNODES;                        // N*64
    float* bufB = bufA + NNODES * DIM;                // N*64

    const int B = 256;
    const int nS  = KFAC * NEDGES;                    // 2,000,000
    const int nNE = NNODES * DIM;                     // 6,400,000

    // init: S working copy, degree -> dinv, ego + all_embs seed
    k_copy<<<(nS + B - 1) / B, B, 0, stream>>>(S_in, Sw, nS);
    k_zero<<<(NNODES + B - 1) / B, B, 0, stream>>>(dinv, NNODES);
    k_deg<<<(NEDGES + B - 1) / B, B, 0, stream>>>(row, col, dinv);
    k_dinv<<<(NNODES + B - 1) / B, B, 0, stream>>>(dinv);
    k_init_ego<<<nNE / B, B, 0, stream>>>(user_emb, item_emb, bufA, all);

    const float* ego = bufA;
    float* xnew = bufB;
    const int convBlocks  = (2 * NEDGES * DIM) / B;   // 250,000 exact
    const int routeBlocks = (NEDGES / 16) * KFAC / 8; // 15,625 exact

    for (int layer = 0; layer < 2; ++layer) {
        for (int it = 0; it < 2; ++it) {
            k_softmax<<<(NEDGES + B - 1) / B, B, 0, stream>>>(Sw);
            k_zero<<<nNE / B, B, 0, stream>>>(xnew, nNE);
            k_conv<<<convBlocks, B, 0, stream>>>(row, col, dinv, Sw, ego, xnew);
            k_routing<<<routeBlocks, B, 0, stream>>>(row, col, ego, xnew, Sw);
        }
        k_accum<<<nNE / B, B, 0, stream>>>(all, xnew);
        const float* t = ego; ego = xnew; xnew = (float*)t;   // ping-pong
    }
}